// Rank1Canvas_12463995093171
// MI455X (gfx1250) — compile-verified
//
#include <hip/hip_runtime.h>

typedef __attribute__((ext_vector_type(2))) float v2f;
typedef __attribute__((ext_vector_type(8))) float v8f;

#define NN 512
#define HW (NN * NN)          // 262144
#define BATCH 32
#define KPAD 34               // K-slab row pitch (even -> 8B aligned pairs)

// ---------------------------------------------------------------------------
// Kernel 1: luma = 0.299 R + 0.587 G + 0.114 B   (pure streaming, float4)
// ---------------------------------------------------------------------------
__global__ __launch_bounds__(256) void luma_kernel(const float* __restrict__ x,
                                                   float* __restrict__ y) {
    const int g = blockIdx.x * 256 + threadIdx.x;   // float4 group, B*HW/4 total
    const int b = g >> 16;                          // HW/4 = 65536 groups/sample
    const int p = g & 65535;
    const float4 r  = ((const float4*)(x + ((size_t)b * 3 + 0) * HW))[p];
    const float4 gg = ((const float4*)(x + ((size_t)b * 3 + 1) * HW))[p];
    const float4 bb = ((const float4*)(x + ((size_t)b * 3 + 2) * HW))[p];
    float4 o;
    o.x = 0.299f * r.x + 0.587f * gg.x + 0.114f * bb.x;
    o.y = 0.299f * r.y + 0.587f * gg.y + 0.114f * bb.y;
    o.z = 0.299f * r.z + 0.587f * gg.z + 0.114f * bb.z;
    o.w = 0.299f * r.w + 0.587f * gg.w + 0.114f * bb.w;
    ((float4*)y)[g] = o;
}

// ---------------------------------------------------------------------------
// Kernel 2: Gram G = Y^T Y per sample, via V_WMMA_F32_16X16X4_F32.
// WG = 256 threads (8 waves). Each wave owns one 16x16 tile of G;
// the WG covers a 64(i) x 32(j) tile. K staged through LDS 32 deep,
// K-contiguous layout -> fragment reads lower to ds_load_2addr_b64.
// Staging uses b128 global loads + statically unrolled transposed stores.
// ---------------------------------------------------------------------------
__global__ __launch_bounds__(256) void gram_kernel(const float* __restrict__ Y,
                                                   float* __restrict__ G) {
    __shared__ __align__(16) float sA[64 * KPAD];   // [i][k]: A columns (i-range)
    __shared__ __align__(16) float sB[32 * KPAD];   // [j][k]: B columns (j-range)

    const int t    = threadIdx.x;
    const int b    = blockIdx.z;
    const int i0   = blockIdx.y * 64;
    const int j0   = blockIdx.x * 32;
    const float* Yb = Y + (size_t)b * HW;

    const int lane = t & 31;
    const int wave = t >> 5;
    const int wi   = wave & 3;     // i sub-tile (0..3)
    const int wj   = wave >> 2;    // j sub-tile (0..1)
    const int m    = lane & 15;
    const int half = lane >> 4;    // ISA layout: lanes16-31 hold K=2,3

    const float* aRow = sA + (wi * 16 + m) * KPAD;  // this lane's A column
    const float* bRow = sB + (wj * 16 + m) * KPAD;  // this lane's B column

    // Precomputed staging coordinates (constant across K-slabs)
    const int akA = t >> 4;              // sA: 16 float4 per k-row
    const int aiA = (t & 15) * 4;
    const int akB = t >> 3;              // sB: 8 float4 per k-row
    const int ajB = (t & 7) * 4;

    v8f acc = {};                  // 16x16 f32 accumulator (8 VGPRs)

    for (int k0 = 0; k0 < NN; k0 += 32) {
        // --- stage sA: 32(k) x 64(i) = 512 float4; 2 per thread ---
#pragma unroll
        for (int s = 0; s < 2; ++s) {
            const int k  = akA + s * 16;
            const float4 d =
                *(const float4*)(Yb + (size_t)(k0 + k) * NN + (i0 + aiA));
            float* p = sA + aiA * KPAD + k;
            p[0 * KPAD] = d.x;
            p[1 * KPAD] = d.y;
            p[2 * KPAD] = d.z;
            p[3 * KPAD] = d.w;
        }
        // --- stage sB: 32(k) x 32(j) = 256 float4; 1 per thread ---
        {
            const float4 d =
                *(const float4*)(Yb + (size_t)(k0 + akB) * NN + (j0 + ajB));
            float* p = sB + ajB * KPAD + akB;
            p[0 * KPAD] = d.x;
            p[1 * KPAD] = d.y;
            p[2 * KPAD] = d.z;
            p[3 * KPAD] = d.w;
        }
        __syncthreads();

#pragma unroll
        for (int kk = 0; kk < 8; ++kk) {
            // A 16x4 layout: VGPR0 = K{0|2}, VGPR1 = K{1|3}, M = lane&15.
            // kbase is always even -> 8-byte-aligned v2f (ds_load_b64 pairs).
            const int kbase = kk * 4 + half * 2;
            const v2f a  = *(const v2f*)(aRow + kbase);
            const v2f bm = *(const v2f*)(bRow + kbase);
            acc = __builtin_amdgcn_wmma_f32_16x16x4_f32(
                false, a, false, bm, (short)0, acc, false, false);
        }
        __syncthreads();
    }

    // C/D layout: VGPR r -> row M = r (lanes 0-15) / r+8 (lanes 16-31), N = lane&15
    const int col = j0 + wj * 16 + m;
    float* Gb = G + (size_t)b * HW;
#pragma unroll
    for (int r = 0; r < 8; ++r) {
        const int row = i0 + wi * 16 + r + half * 8;
        Gb[(size_t)row * NN + col] = acc[r];
    }
}

// ---------------------------------------------------------------------------
// Kernel 3: 256 power iterations on G (squared convergence vs Y-matvecs),
// then w = Y v and coalesced outer-product write  out = w v^T.
// One 512-thread WG (16 wave32) per sample.
// ---------------------------------------------------------------------------
__global__ __launch_bounds__(512) void powiter_kernel(const float* __restrict__ Y,
                                                      const float* __restrict__ G,
                                                      float* __restrict__ out) {
    __shared__ float v[NN];
    __shared__ float nv[NN];
    __shared__ float red[16];

    const int t = threadIdx.x;
    const int b = blockIdx.x;
    const float* Gb = G + (size_t)b * HW;
    const float* Yb = Y + (size_t)b * HW;

    // deterministic jittered init (avoid measure-zero orthogonal start)
    const unsigned h = (unsigned)t * 2654435761u + 0x9e3779b9u;
    v[t] = 1.0f + 9.765625e-4f * (float)(h & 1023u);
    __syncthreads();

    for (int it = 0; it < 256; ++it) {
        const float4* row = (const float4*)(Gb + (size_t)t * NN);
        float acc = 0.0f;
#pragma unroll 4
        for (int j4 = 0; j4 < NN / 4; ++j4) {
            const float4 g4 = row[j4];
            acc += g4.x * v[j4 * 4 + 0] + g4.y * v[j4 * 4 + 1] +
                   g4.z * v[j4 * 4 + 2] + g4.w * v[j4 * 4 + 3];
        }
        nv[t] = acc;

        float s = acc * acc;                       // wave32 reduction
        for (int off = 16; off > 0; off >>= 1)
            s += __shfl_down(s, off, 32);
        if ((t & 31) == 0) red[t >> 5] = s;
        __syncthreads();

        float tot = 0.0f;
#pragma unroll
        for (int wv = 0; wv < 16; ++wv) tot += red[wv];
        const float newv = nv[t] * rsqrtf(tot);
        __syncthreads();                           // red reads done before reuse
        v[t] = newv;
        __syncthreads();
    }

    // w = Y v  (row t of Y is contiguous)
    {
        const float4* row = (const float4*)(Yb + (size_t)t * NN);
        float acc = 0.0f;
#pragma unroll 4
        for (int j4 = 0; j4 < NN / 4; ++j4) {
            const float4 y4 = row[j4];
            acc += y4.x * v[j4 * 4 + 0] + y4.y * v[j4 * 4 + 1] +
                   y4.z * v[j4 * 4 + 2] + y4.w * v[j4 * 4 + 3];
        }
        nv[t] = acc;    // nv now holds w = sigma0 * u0
    }
    __syncthreads();

    // out[b,i,j] = w[i] * v[j], float4 coalesced
    float4* ob = (float4*)(out + (size_t)b * HW);
    for (int g = t; g < HW / 4; g += 512) {
        const int i  = g >> 7;             // 128 float4 per row
        const int j4 = (g & 127) * 4;
        const float wi_ = nv[i];
        float4 o;
        o.x = wi_ * v[j4 + 0];
        o.y = wi_ * v[j4 + 1];
        o.z = wi_ * v[j4 + 2];
        o.w = wi_ * v[j4 + 3];
        ob[g] = o;
    }
}

// ---------------------------------------------------------------------------
extern "C" void kernel_launch(void* const* d_in, const int* in_sizes, int n_in,
                              void* d_out, int out_size, void* d_ws, size_t ws_size,
                              hipStream_t stream) {
    (void)in_sizes; (void)n_in; (void)out_size; (void)ws_size;
    const float* x = (const float*)d_in[0];
    float* out = (float*)d_out;

    float* Yl = (float*)d_ws;                       // 32 MB luma
    float* G  = Yl + (size_t)BATCH * HW;            // 32 MB Gram

    luma_kernel<<<(BATCH * HW / 4) / 256, 256, 0, stream>>>(x, Yl);

    dim3 gg(NN / 32, NN / 64, BATCH);               // (j tiles, i tiles, batch)
    gram_kernel<<<gg, 256, 0, stream>>>(Yl, G);

    powiter_kernel<<<BATCH, 512, 0, stream>>>(Yl, G, out);
}